// SAGEConv_54614804136338
// MI455X (gfx1250) — compile-verified
//
#include <hip/hip_runtime.h>

// SAGEConv: mean-aggregate neighbors, concat with self, dense [256->256] + bias.
// N=10000 nodes, E=640000 edges, IN=128, OUT=256.

#define N_NODES 10000
#define N_EDGES 640000
#define IN_FEAT 128
#define OUT_FEAT 256

typedef __attribute__((ext_vector_type(2))) float v2f;
typedef __attribute__((ext_vector_type(8))) float v8f;

// ---------------------------------------------------------------------------
// Kernel 1: zero accumulator + degree (ws is poisoned; must re-init per call)
// ---------------------------------------------------------------------------
__global__ void sage_zero(float* __restrict__ p, int n) {
    int i = blockIdx.x * blockDim.x + threadIdx.x;
    if (i < n) p[i] = 0.0f;
}

// ---------------------------------------------------------------------------
// Kernel 2: edge scatter. One thread per (edge, feature). A wave32 covers 32
// consecutive features of one edge -> coalesced 128B gather from h[src] and
// coalesced global_atomic_add_f32 into acc[dst]. All traffic is L2-resident.
// ---------------------------------------------------------------------------
__global__ void sage_scatter(const float* __restrict__ h,
                             const int*   __restrict__ src,
                             const int*   __restrict__ dst,
                             float* __restrict__ acc,
                             float* __restrict__ deg) {
    long long i = (long long)blockIdx.x * blockDim.x + threadIdx.x;
    if (i >= (long long)N_EDGES * IN_FEAT) return;
    int f = (int)(i & (IN_FEAT - 1));
    int e = (int)(i >> 7);
    int s = src[e];
    int d = dst[e];
    atomicAdd(&acc[(long long)d * IN_FEAT + f], h[(long long)s * IN_FEAT + f]);
    if (f == 0) atomicAdd(&deg[d], 1.0f);
}

// ---------------------------------------------------------------------------
// Kernel 3: neighbor mean.  hN = deg>0 ? acc/deg : 0
// ---------------------------------------------------------------------------
__global__ void sage_mean(const float* __restrict__ acc,
                          const float* __restrict__ deg,
                          float* __restrict__ hN) {
    int i = blockIdx.x * blockDim.x + threadIdx.x;
    if (i < N_NODES * IN_FEAT) {
        float d = deg[i >> 7];
        hN[i] = (d > 0.0f) ? acc[i] / d : 0.0f;
    }
}

// ---------------------------------------------------------------------------
// Kernel 4: out = [h | hN] @ W + b   via V_WMMA_F32_16X16X4_F32 (exact f32).
// One wave per 16x16 output tile; 625 x 16 = 10000 tiles, 8 waves/block.
//
// A frag (16x4 f32, 2 VGPRs): lane&15 = row M; lanes 0-15 hold K,K+1,
//                             lanes 16-31 hold K+2,K+3.
// B frag (4x16 f32, 2 VGPRs): lane&15 = col N; same K split as A.
// C/D (16x16 f32, 8 VGPRs):   VGPR v -> row v + 8*(lane>=16), col lane&15.
// ---------------------------------------------------------------------------
__global__ void __launch_bounds__(256)
sage_gemm_wmma(const float* __restrict__ h,
               const float* __restrict__ hN,
               const float* __restrict__ W,
               const float* __restrict__ b,
               float* __restrict__ out) {
    const int lane  = threadIdx.x & 31;
    const int wave  = threadIdx.x >> 5;
    const int tile  = blockIdx.x * 8 + wave;     // 0..9999, exact fit
    const int mTile = tile >> 4;                 // 0..624
    const int nTile = tile & 15;                 // 0..15
    const int m0 = mTile * 16;
    const int n0 = nTile * 16;

    const int nl    = lane & 15;                 // col within tile (B/C/D) / row (A)
    const int khalf = (lane >> 4) << 1;          // 0 or 2: K sub-offset in frags

    // Seed accumulator with the bias (depends on column only).
    const float bias = b[n0 + nl];
    v8f c;
#pragma unroll
    for (int v = 0; v < 8; ++v) c[v] = bias;

    // ---- K = 0..127 : self features (h) -----------------------------------
    const float* __restrict__ Arow = h + (long long)(m0 + nl) * IN_FEAT;
#pragma unroll 8
    for (int k = 0; k < IN_FEAT; k += 4) {
        const int kb = k + khalf;
        v2f a;
        a.x = Arow[kb];
        a.y = Arow[kb + 1];
        v2f bb;
        bb.x = W[(long long)kb * OUT_FEAT + n0 + nl];
        bb.y = W[(long long)(kb + 1) * OUT_FEAT + n0 + nl];
        c = __builtin_amdgcn_wmma_f32_16x16x4_f32(
                /*neg_a=*/false, a, /*neg_b=*/false, bb,
                /*c_mod=*/(short)0, c, /*reuse_a=*/false, /*reuse_b=*/false);
    }

    // ---- K = 128..255 : neighbor-mean features (hN) -----------------------
    const float* __restrict__ Nrow = hN + (long long)(m0 + nl) * IN_FEAT;
#pragma unroll 8
    for (int k = 0; k < IN_FEAT; k += 4) {
        const int kb = k + khalf;
        const int kw = IN_FEAT + kb;             // row in W
        v2f a;
        a.x = Nrow[kb];
        a.y = Nrow[kb + 1];
        v2f bb;
        bb.x = W[(long long)kw * OUT_FEAT + n0 + nl];
        bb.y = W[(long long)(kw + 1) * OUT_FEAT + n0 + nl];
        c = __builtin_amdgcn_wmma_f32_16x16x4_f32(
                false, a, false, bb, (short)0, c, false, false);
    }

    // ---- Store: VGPR v holds (row = m0 + v + 8*(lane>>4), col = n0 + nl) ---
    const int mrow = m0 + ((lane >> 4) << 3);
#pragma unroll
    for (int v = 0; v < 8; ++v)
        out[(long long)(mrow + v) * OUT_FEAT + n0 + nl] = c[v];
}

// ---------------------------------------------------------------------------
// Launch
// ---------------------------------------------------------------------------
extern "C" void kernel_launch(void* const* d_in, const int* in_sizes, int n_in,
                              void* d_out, int out_size, void* d_ws, size_t ws_size,
                              hipStream_t stream) {
    const float* h   = (const float*)d_in[0];
    const int*   src = (const int*)  d_in[1];
    const int*   dst = (const int*)  d_in[2];
    const float* W   = (const float*)d_in[3];
    const float* b   = (const float*)d_in[4];
    float*       out = (float*)d_out;

    // Workspace layout: acc [N*IN] | deg [N] | hN [N*IN]  (~10.3 MB)
    float* acc = (float*)d_ws;
    float* deg = acc + (size_t)N_NODES * IN_FEAT;
    float* hN  = deg + N_NODES;

    // 1) zero acc+deg (contiguous)
    const int zn = N_NODES * IN_FEAT + N_NODES;
    sage_zero<<<(zn + 255) / 256, 256, 0, stream>>>(acc, zn);

    // 2) edge scatter-sum + degree
    const long long sthreads = (long long)N_EDGES * IN_FEAT;
    sage_scatter<<<(int)((sthreads + 255) / 256), 256, 0, stream>>>(h, src, dst, acc, deg);

    // 3) mean
    const int mn = N_NODES * IN_FEAT;
    sage_mean<<<(mn + 255) / 256, 256, 0, stream>>>(acc, deg, hN);

    // 4) WMMA GEMM: 10000 tiles / 8 waves per block = 1250 blocks
    sage_gemm_wmma<<<1250, 256, 0, stream>>>(h, hN, W, b, out);
}